// CorrectionByf3d_31963146616886
// MI455X (gfx1250) — compile-verified
//
#include <hip/hip_runtime.h>

// ---------------------------------------------------------------------------
// Problem constants (match the JAX reference)
// ---------------------------------------------------------------------------
#define N_POINTS    262144
#define N_VERTS     6890
#define N_JOINTS    24
#define TILES_TOTAL ((N_VERTS + 15) / 16)                 // 431 vertex tiles
#define CHUNK_TILES 128                                   // 2048 verts -> 32KB LDS
#define N_CHUNKS    ((TILES_TOTAL + CHUNK_TILES - 1) / CHUNK_TILES)  // 4

typedef __attribute__((ext_vector_type(2))) float v2f;
typedef __attribute__((ext_vector_type(8))) float v8f;

// ---------------------------------------------------------------------------
// Kernel 1: fused KNN argmin via V_WMMA_F32_16X16X4_F32
//   score(q, v) = |v|^2 - 2 q.v   (|q|^2 dropped: constant per row)
//   A row (16x4):  [-2qx, -2qy, -2qz, 1]
//   B col (4x16):  [ vx,   vy,   vz, |v|^2]
//   => D = score, argmin over columns fused in registers.
// A-operand f32 16x4 layout: lanes 0-15 -> {K0,K1}, lanes 16-31 -> {K2,K3}.
// B-operand f32 4x16 layout: lane = column N (mod 16), same K split.
// C/D 16x16 f32: lane L<16 -> rows M=0..7 (VGPR j -> M=j) at col N=L,
//                lane L>=16 -> rows M=8..15 at col N=L-16.
// ---------------------------------------------------------------------------
__global__ __launch_bounds__(256) void knn_wmma_kernel(
    const float* __restrict__ q, const float* __restrict__ verts,
    int* __restrict__ idxOut)
{
  __shared__ float lds[CHUNK_TILES * 16 * 4];  // {x,y,z,|v|^2} per vertex

  const int lane  = threadIdx.x & 31;
  const int wave  = threadIdx.x >> 5;
  const int col   = lane & 15;                 // column N owned by this lane
  const int qBase = blockIdx.x * 128 + wave * 16;
  const int row   = qBase + col;

  const float qx = q[3 * row + 0];
  const float qy = q[3 * row + 1];
  const float qz = q[3 * row + 2];

  v2f a;
  if (lane < 16) { a.x = -2.0f * qx; a.y = -2.0f * qy; }  // K=0,1
  else           { a.x = -2.0f * qz; a.y = 1.0f;       }  // K=2,3

  float best[8];
  int   btile[8];
#pragma unroll
  for (int j = 0; j < 8; ++j) { best[j] = 3.0e38f; btile[j] = 0; }

  for (int chunk = 0; chunk < N_CHUNKS; ++chunk) {
    const int tile0  = chunk * CHUNK_TILES;
    const int ntiles = min(CHUNK_TILES, TILES_TOTAL - tile0);
    const int nv     = ntiles * 16;

    // Cooperative stage of this vertex chunk into LDS as {x,y,z,|v|^2}.
    for (int e = threadIdx.x; e < nv; e += 256) {
      const int gv = tile0 * 16 + e;
      float x = 0.0f, y = 0.0f, z = 0.0f, v2 = 3.0e38f;  // pad: never selected
      if (gv < N_VERTS) {
        x = verts[3 * gv + 0];
        y = verts[3 * gv + 1];
        z = verts[3 * gv + 2];
        v2 = x * x + y * y + z * z;
      }
      lds[4 * e + 0] = x;
      lds[4 * e + 1] = y;
      lds[4 * e + 2] = z;
      lds[4 * e + 3] = v2;
    }
    __syncthreads();

    const float* bp = &lds[4 * col + ((lane < 16) ? 0 : 2)];
    for (int t = 0; t < ntiles; ++t) {
      v2f b = *(const v2f*)(bp + t * 64);      // ds_load_b64, conflict-free
      v8f c = {};
      v8f d = __builtin_amdgcn_wmma_f32_16x16x4_f32(
          /*neg_a=*/false, a, /*neg_b=*/false, b,
          /*c_mod=*/(short)0, c, /*reuse_a=*/false, /*reuse_b=*/false);
      const int tileId = tile0 + t;
#pragma unroll
      for (int j = 0; j < 8; ++j) {
        const bool take = d[j] < best[j];      // strict <: keeps lowest tile id
        best[j]  = take ? d[j]   : best[j];
        btile[j] = take ? tileId : btile[j];
      }
    }
    __syncthreads();
  }

  // Cross-lane argmin over the 16 columns of each half-wave (rows split
  // M=0..7 on lanes 0-15, M=8..15 on lanes 16-31). Tie-break on lowest
  // vertex index to match jnp.argmin first-hit semantics.
  int vidx[8];
#pragma unroll
  for (int j = 0; j < 8; ++j) vidx[j] = btile[j] * 16 + col;

#pragma unroll
  for (int m = 1; m < 16; m <<= 1) {
#pragma unroll
    for (int j = 0; j < 8; ++j) {
      const float ob = __shfl_xor(best[j], m, 32);
      const int   oi = __shfl_xor(vidx[j], m, 32);
      const bool take = (ob < best[j]) || ((ob == best[j]) && (oi < vidx[j]));
      best[j] = take ? ob : best[j];
      vidx[j] = take ? oi : vidx[j];
    }
  }

  if (col == 0) {
    const int mBase = qBase + ((lane >= 16) ? 8 : 0);
#pragma unroll
    for (int j = 0; j < 8; ++j) idxOut[mBase + j] = vidx[j];
  }
}

// ---------------------------------------------------------------------------
// Kernel 2: per-point blend-skinning tail.
//   bw = weights[idx]; A = bw . A_t; A2 = bw . A_big (only rows 0..2 needed)
//   can = inv(A[:3,:3]) (q - A[:3,3]);  out = A2[:3,:3] can + A2[:3,3]
// 3x3 inverse via adjugate / determinant (mathematically == linalg.inv).
// ---------------------------------------------------------------------------
__global__ __launch_bounds__(256) void skin_kernel(
    const float* __restrict__ q, const float* __restrict__ weights,
    const float* __restrict__ At, const float* __restrict__ Ab,
    const int* __restrict__ idx, float* __restrict__ out)
{
  __shared__ float sA[N_JOINTS * 16];
  __shared__ float sB[N_JOINTS * 16];
  for (int i = threadIdx.x; i < N_JOINTS * 16; i += 256) {
    sA[i] = At[i];
    sB[i] = Ab[i];
  }
  __syncthreads();

  const int n = blockIdx.x * 256 + threadIdx.x;
  const int v = idx[n];
  const float* __restrict__ w = weights + v * N_JOINTS;

  float A[12], B[12];
#pragma unroll
  for (int i = 0; i < 12; ++i) { A[i] = 0.0f; B[i] = 0.0f; }

  for (int k = 0; k < N_JOINTS; ++k) {
    const float wk = w[k];
#pragma unroll
    for (int i = 0; i < 12; ++i) {
      A[i] = fmaf(wk, sA[k * 16 + i], A[i]);
      B[i] = fmaf(wk, sB[k * 16 + i], B[i]);
    }
  }

  const float px = q[3 * n + 0], py = q[3 * n + 1], pz = q[3 * n + 2];
  const float cx = px - A[3], cy = py - A[7], cz = pz - A[11];

  const float m00 = A[0], m01 = A[1], m02 = A[2];
  const float m10 = A[4], m11 = A[5], m12 = A[6];
  const float m20 = A[8], m21 = A[9], m22 = A[10];

  const float c00 = m11 * m22 - m12 * m21;
  const float c01 = m12 * m20 - m10 * m22;
  const float c02 = m10 * m21 - m11 * m20;
  const float det = m00 * c00 + m01 * c01 + m02 * c02;
  const float rd  = 1.0f / det;

  const float i00 = c00 * rd;
  const float i01 = (m02 * m21 - m01 * m22) * rd;
  const float i02 = (m01 * m12 - m02 * m11) * rd;
  const float i10 = c01 * rd;
  const float i11 = (m00 * m22 - m02 * m20) * rd;
  const float i12 = (m02 * m10 - m00 * m12) * rd;
  const float i20 = c02 * rd;
  const float i21 = (m01 * m20 - m00 * m21) * rd;
  const float i22 = (m00 * m11 - m01 * m10) * rd;

  const float ux = i00 * cx + i01 * cy + i02 * cz;
  const float uy = i10 * cx + i11 * cy + i12 * cz;
  const float uz = i20 * cx + i21 * cy + i22 * cz;

  out[3 * n + 0] = B[0] * ux + B[1] * uy + B[2]  * uz + B[3];
  out[3 * n + 1] = B[4] * ux + B[5] * uy + B[6]  * uz + B[7];
  out[3 * n + 2] = B[8] * ux + B[9] * uy + B[10] * uz + B[11];
}

// ---------------------------------------------------------------------------
// Launch
// ---------------------------------------------------------------------------
extern "C" void kernel_launch(void* const* d_in, const int* in_sizes, int n_in,
                              void* d_out, int out_size, void* d_ws, size_t ws_size,
                              hipStream_t stream) {
  const float* q       = (const float*)d_in[0];  // [262144,3]
  const float* verts   = (const float*)d_in[1];  // [6890,3]
  const float* weights = (const float*)d_in[2];  // [6890,24]
  const float* At      = (const float*)d_in[3];  // [24,4,4]
  const float* Ab      = (const float*)d_in[4];  // [24,4,4]
  float* out = (float*)d_out;                    // [262144,3]
  int* idxbuf = (int*)d_ws;                      // N_POINTS ints of scratch

  (void)in_sizes; (void)n_in; (void)out_size; (void)ws_size;

  // 2048 blocks x 256 threads: each block = 128 query rows (8 waves x 16).
  knn_wmma_kernel<<<N_POINTS / 128, 256, 0, stream>>>(q, verts, idxbuf);
  // 1024 blocks x 256 threads: 1 thread per point.
  skin_kernel<<<N_POINTS / 256, 256, 0, stream>>>(q, weights, At, Ab, idxbuf, out);
}